// GNNModel_3882650436959
// MI455X (gfx1250) — compile-verified
//
#include <hip/hip_runtime.h>
#include <hip/hip_bf16.h>
#include <math.h>

typedef float v2f __attribute__((ext_vector_type(2)));
typedef float v8f __attribute__((ext_vector_type(8)));

#define D_IN 128
#define D_H  64
#define D_OUT 32

// ---------------------------------------------------------------------------
// Workspace init: deg = 1.0 (self loop), agg1 = 0, agg2 = 0, gsum = 0
// Region map (in floats): [0,N) deg=1 ; [N,65N) agg1=0 ; [65N,97N) agg2=0 ;
// [97N,97N+32) gsum=0
// ---------------------------------------------------------------------------
__global__ void init_ws_kernel(float* __restrict__ deg,
                               float* __restrict__ agg1,
                               float* __restrict__ agg2,
                               float* __restrict__ gsum,
                               long long N) {
    long long gid = (long long)blockIdx.x * blockDim.x + threadIdx.x;
    if (gid < N) {
        deg[gid] = 1.0f;
    } else if (gid < 65LL * N) {
        agg1[gid - N] = 0.0f;
    } else if (gid < 97LL * N) {
        agg2[gid - 65LL * N] = 0.0f;
    } else if (gid < 97LL * N + 32) {
        gsum[gid - 97LL * N] = 0.0f;
    }
}

// ---------------------------------------------------------------------------
// In-degree accumulation over edges (A + I handled via deg init = 1.0)
// ---------------------------------------------------------------------------
__global__ void degree_kernel(const int* __restrict__ dst,
                              float* __restrict__ deg, int E) {
    int e = blockIdx.x * blockDim.x + threadIdx.x;
    if (e < E) atomicAdd(deg + dst[e], 1.0f);
}

__global__ void dinv_kernel(const float* __restrict__ deg,
                            float* __restrict__ dinv, int N) {
    int i = blockIdx.x * blockDim.x + threadIdx.x;
    if (i < N) dinv[i] = rsqrtf(deg[i]);
}

// ---------------------------------------------------------------------------
// f32 GEMM via V_WMMA_F32_16X16X4_F32.  One 16x16 output tile per wave32.
// A: MxK row-major, B: KxN row-major, C: MxN row-major.
// Requires M%16==0, K%4==0, N%16==0 (true here: 100000x128x64 and x64x32).
//
// Per-K-step-of-4 operand layout (wave32):
//   A 16x4:  lanes 0-15 -> rows, VGPR{0,1} = K {k,k+1}; lanes 16-31 = K {k+2,k+3}
//   B 4x16:  lanes 0-15 -> cols, VGPR{0,1} = K {k,k+1}; lanes 16-31 = K {k+2,k+3}
//   C 16x16: VGPR r: lanes 0-15 -> M=r, lanes 16-31 -> M=r+8, N = lane&15
// ---------------------------------------------------------------------------
__global__ void gemm_wmma_f32_kernel(const float* __restrict__ A,
                                     const float* __restrict__ B,
                                     float* __restrict__ C,
                                     int M, int K, int Ncols) {
    const int lane = threadIdx.x & 31;
    const int wave = threadIdx.x >> 5;
    const int tilesN = Ncols >> 4;
    const int tile = blockIdx.x * (blockDim.x >> 5) + wave;
    const int tileM = tile / tilesN;
    const int tileN = tile - tileM * tilesN;
    if (tileM * 16 >= M) return;              // wave-uniform: EXEC stays all-1s

    const int mrow = tileM * 16 + (lane & 15);
    const int col  = tileN * 16 + (lane & 15);
    const int koff = (lane >> 4) << 1;        // 0 for lanes 0-15, 2 for 16-31

    v8f c = {};
    for (int k = 0; k < K; k += 4) {
        v2f a = *(const v2f*)(A + (size_t)mrow * K + (k + koff));
        v2f b;
        b.x = B[(size_t)(k + koff)     * Ncols + col];
        b.y = B[(size_t)(k + koff + 1) * Ncols + col];
        c = __builtin_amdgcn_wmma_f32_16x16x4_f32(
                /*neg_a=*/false, a, /*neg_b=*/false, b,
                /*c_mod=*/(short)0, c, /*reuse_a=*/false, /*reuse_b=*/false);
    }

    const int rowbase = tileM * 16 + ((lane >> 4) << 3);  // +8 for upper lanes
#pragma unroll
    for (int r = 0; r < 8; ++r) {
        C[(size_t)(rowbase + r) * Ncols + col] = c[r];
    }
}

// ---------------------------------------------------------------------------
// Edge scatter: agg[dst] += h[src] * dinv[src] * dinv[dst]
// One thread per (edge, 4-feature group); float4 gather (L2-resident h).
// gshift = log2(feats/4)
// ---------------------------------------------------------------------------
__global__ void scatter_kernel(const int* __restrict__ src,
                               const int* __restrict__ dst,
                               const float* __restrict__ dinv,
                               const float* __restrict__ h,
                               float* __restrict__ agg,
                               long long E, int feats, int gshift) {
    long long gid = (long long)blockIdx.x * blockDim.x + threadIdx.x;
    long long total = E << gshift;
    if (gid >= total) return;
    int e  = (int)(gid >> gshift);
    int fb = ((int)gid & ((1 << gshift) - 1)) << 2;
    int s = src[e], d = dst[e];
    float nrm = dinv[s] * dinv[d];
    const float4 v = *(const float4*)(h + (size_t)s * feats + fb);
    float* ap = agg + (size_t)d * feats + fb;
    atomicAdd(ap + 0, v.x * nrm);
    atomicAdd(ap + 1, v.y * nrm);
    atomicAdd(ap + 2, v.z * nrm);
    atomicAdd(ap + 3, v.w * nrm);
}

// ---------------------------------------------------------------------------
// Self-loop + bias + ReLU, in place into agg:  agg = relu(agg + h/deg + b)
// fshift = log2(feats)
// ---------------------------------------------------------------------------
__global__ void selfloop_relu_kernel(const float* __restrict__ h,
                                     const float* __restrict__ dinv,
                                     const float* __restrict__ bias,
                                     float* __restrict__ agg,
                                     long long N, int feats, int fshift) {
    long long gid = (long long)blockIdx.x * blockDim.x + threadIdx.x;
    if (gid >= N * feats) return;
    int node = (int)(gid >> fshift);
    int f    = (int)gid & (feats - 1);
    float di = dinv[node];
    float v = agg[gid] + h[gid] * di * di + bias[f];
    agg[gid] = v > 0.0f ? v : 0.0f;
}

// ---------------------------------------------------------------------------
// Fused layer-2 self-loop + ReLU + column-sum readout into gsum[32].
// Block: 256 threads = 8 node-lanes x 32 features, LDS reduce, 32 atomics/blk.
// ---------------------------------------------------------------------------
__global__ void readout_kernel(const float* __restrict__ h2,
                               const float* __restrict__ agg2,
                               const float* __restrict__ dinv,
                               const float* __restrict__ b2,
                               float* __restrict__ gsum, int N) {
    __shared__ float sm[256];
    int f = threadIdx.x & 31;
    int r = threadIdx.x >> 5;   // 0..7
    float acc = 0.0f;
    for (int i = blockIdx.x * 8 + r; i < N; i += gridDim.x * 8) {
        float di = dinv[i];
        size_t idx = (size_t)i * 32 + f;
        float v = agg2[idx] + h2[idx] * di * di + b2[f];
        acc += v > 0.0f ? v : 0.0f;
    }
    sm[threadIdx.x] = acc;
    __syncthreads();
    if (threadIdx.x < 32) {
        float t = 0.0f;
#pragma unroll
        for (int rr = 0; rr < 8; ++rr) t += sm[rr * 32 + f];
        atomicAdd(gsum + f, t);
    }
}

// ---------------------------------------------------------------------------
// Final: out = sigmoid((gsum/N) @ Wfc + bfc)
// ---------------------------------------------------------------------------
__global__ void final_fc_kernel(const float* __restrict__ gsum,
                                const float* __restrict__ Wfc,
                                const float* __restrict__ bfc,
                                float* __restrict__ out, float invN) {
    if (threadIdx.x == 0) {
        float z = 0.0f;
        for (int f = 0; f < 32; ++f) z += gsum[f] * invN * Wfc[f];
        z += bfc[0];
        out[0] = 1.0f / (1.0f + __expf(-z));
    }
}

// ---------------------------------------------------------------------------
extern "C" void kernel_launch(void* const* d_in, const int* in_sizes, int n_in,
                              void* d_out, int out_size, void* d_ws, size_t ws_size,
                              hipStream_t stream) {
    const float* x   = (const float*)d_in[0];
    const int*   ei  = (const int*)d_in[1];
    const float* W1  = (const float*)d_in[2];
    const float* b1  = (const float*)d_in[3];
    const float* W2  = (const float*)d_in[4];
    const float* b2  = (const float*)d_in[5];
    const float* Wfc = (const float*)d_in[6];
    const float* bfc = (const float*)d_in[7];
    float* out = (float*)d_out;

    const long long N = in_sizes[0] / D_IN;     // 100000
    const long long E = in_sizes[1] / 2;        // 3200000
    const int* src = ei;
    const int* dst = ei + E;

    // Workspace layout (floats)
    float* ws   = (float*)d_ws;
    float* deg  = ws;                  // N
    float* dinv = deg  + N;            // N
    float* h1   = dinv + N;            // N*64
    float* agg1 = h1   + N * D_H;      // N*64
    float* h2   = agg1 + N * D_H;      // N*32
    float* agg2 = h2   + N * D_OUT;    // N*32
    float* gsum = agg2 + N * D_OUT;    // 32

    const int TB = 256;

    // 1) init deg/agg1/agg2/gsum
    {
        long long total = 97LL * N + 32;
        int blocks = (int)((total + TB - 1) / TB);
        init_ws_kernel<<<blocks, TB, 0, stream>>>(deg, agg1, agg2, gsum, N);
    }
    // 2) degrees
    degree_kernel<<<(int)((E + TB - 1) / TB), TB, 0, stream>>>(dst, deg, (int)E);
    // 3) dinv
    dinv_kernel<<<(int)((N + TB - 1) / TB), TB, 0, stream>>>(deg, dinv, (int)N);
    // 4) GEMM1: h1 = x @ W1   (N x 128 x 64), one tile per wave, 4 waves/block
    {
        int tiles = (int)(N / 16) * (D_H / 16);
        int blocks = (tiles + 3) / 4;
        gemm_wmma_f32_kernel<<<blocks, 128, 0, stream>>>(x, W1, h1, (int)N, D_IN, D_H);
    }
    // 5) scatter layer 1: 16 groups of 4 features per edge
    {
        long long total = E << 4;
        scatter_kernel<<<(int)((total + TB - 1) / TB), TB, 0, stream>>>(
            src, dst, dinv, h1, agg1, E, D_H, 4);
    }
    // 6) self-loop + bias + relu -> agg1 (in place)
    {
        long long total = N * D_H;
        selfloop_relu_kernel<<<(int)((total + TB - 1) / TB), TB, 0, stream>>>(
            h1, dinv, b1, agg1, N, D_H, 6);
    }
    // 7) GEMM2: h2 = agg1 @ W2  (N x 64 x 32)
    {
        int tiles = (int)(N / 16) * (D_OUT / 16);
        int blocks = (tiles + 3) / 4;
        gemm_wmma_f32_kernel<<<blocks, 128, 0, stream>>>(agg1, W2, h2, (int)N, D_H, D_OUT);
    }
    // 8) scatter layer 2: 8 groups of 4 features per edge
    {
        long long total = E << 3;
        scatter_kernel<<<(int)((total + TB - 1) / TB), TB, 0, stream>>>(
            src, dst, dinv, h2, agg2, E, D_OUT, 3);
    }
    // 9) fused self-loop + relu + column-sum readout
    readout_kernel<<<512, TB, 0, stream>>>(h2, agg2, dinv, b2, gsum, (int)N);
    // 10) final fc + sigmoid
    final_fc_kernel<<<1, 32, 0, stream>>>(gsum, Wfc, bfc, out, 1.0f / (float)N);
}